// BriLLMNodeBias_51573967291059
// MI455X (gfx1250) — compile-verified
//
#include <hip/hip_runtime.h>
#include <cmath>

#define DD 64
#define LL 512
#define VV 16000

typedef float v8f __attribute__((ext_vector_type(8)));
typedef float v2f __attribute__((ext_vector_type(2)));
typedef int   v4i __attribute__((ext_vector_type(4)));

#define AS1 __attribute__((address_space(1)))
#define AS3 __attribute__((address_space(3)))

#if __has_builtin(__builtin_amdgcn_global_load_async_to_lds_b128) && __has_builtin(__builtin_amdgcn_s_wait_asynccnt)
#define HAS_ASYNC_LDS 1
#else
#define HAS_ASYNC_LDS 0
#endif

#if __has_builtin(__builtin_amdgcn_wmma_f32_16x16x4_f32)
#define HAS_WMMA4 1
#else
#define HAS_WMMA4 0
#endif

// ---------------------------------------------------------------------------
// Phase 1: sequential scan. One workgroup, 64 threads (2 waves).
// x lives in LDS; W[step_eids[t+1]] is async-prefetched into a double-buffered
// LDS tile while step t computes, hiding HBM latency on the serial chain.
// ---------------------------------------------------------------------------
__global__ __launch_bounds__(64) void brillm_scan(
    const int* __restrict__ ids, const int* __restrict__ step_eids,
    const float* __restrict__ bias_table, const float* __restrict__ W,
    const float* __restrict__ bias, const float* __restrict__ W_shared,
    const float* __restrict__ bias_shared, const float* __restrict__ a,
    const float* __restrict__ gate_p, const float* __restrict__ pe_scale_p,
    const float* __restrict__ PE, float* __restrict__ x_out)
{
  const int tid = threadIdx.x;
  __shared__ float xs[DD];
#if HAS_ASYNC_LDS
  __shared__ float wbuf[2][DD * DD];   // 2 x 16KB double buffer
#endif
  const float gate = gate_p[0];
  const float pe_scale = pe_scale_p[0];

  xs[tid] = bias_table[(size_t)ids[0] * DD + tid] + pe_scale * PE[tid];

#if HAS_ASYNC_LDS
  {
    // preload step 0 matrix into buffer 0: 64 threads x 16 x b128 = 16KB
    char* s = (char*)(W + (size_t)step_eids[0] * (DD * DD)) + tid * 16;
    char* d = (char*)(&wbuf[0][0]) + tid * 16;
#pragma unroll
    for (int i = 0; i < 16; ++i)
      __builtin_amdgcn_global_load_async_to_lds_b128(
          (AS1 v4i*)(s + i * 1024), (AS3 v4i*)(d + i * 1024), 0, 0);
  }
#endif
  __syncthreads();

  for (int t = 0; t < LL - 1; ++t) {
    const int eid = step_eids[t];
#if HAS_ASYNC_LDS
    if (t + 1 < LL - 1) {
      char* s = (char*)(W + (size_t)step_eids[t + 1] * (DD * DD)) + tid * 16;
      char* d = (char*)(&wbuf[(t + 1) & 1][0]) + tid * 16;
#pragma unroll
      for (int i = 0; i < 16; ++i)
        __builtin_amdgcn_global_load_async_to_lds_b128(
            (AS1 v4i*)(s + i * 1024), (AS3 v4i*)(d + i * 1024), 0, 0);
      __builtin_amdgcn_s_wait_asynccnt(16);   // previous batch of 16 has landed
    } else {
      __builtin_amdgcn_s_wait_asynccnt(0);
    }
    __syncthreads();                           // cross-wave LDS visibility
    const float* Wm = &wbuf[t & 1][0];
#else
    const float* Wm = W + (size_t)eid * (DD * DD);
    if (t + 1 < LL - 1) {                      // global_prefetch_b8 next matrix
      const float* Wn = W + (size_t)step_eids[t + 1] * (DD * DD);
      __builtin_prefetch(Wn + tid * 64, 0, 0);
      __builtin_prefetch(Wn + tid * 64 + 32, 0, 0);
    }
#endif
    float acc = 0.f, accs = 0.f;
#pragma unroll 8
    for (int d = 0; d < DD; ++d) {
      const float xd = xs[d];                  // LDS broadcast
      acc  = fmaf(xd, Wm[d * DD + tid], acc);
      accs = fmaf(xd, W_shared[d * DD + tid], accs);  // 16KB, L0-resident
    }
    const int tok = ids[t + 1];
    const float nw = tanhf(acc + bias[(size_t)eid * DD + tid]
                           + pe_scale * PE[(size_t)(t + 1) * DD + tid]
                           + bias_table[(size_t)tok * DD + tid]);
    const float xnew = a[t + 1] * nw + gate * (accs + bias_shared[tid]);
    __syncthreads();
    xs[tid] = xnew;
    __syncthreads();
  }
  x_out[tid] = xs[tid];
}

// ---------------------------------------------------------------------------
// Phase 2: logits. Bandwidth-bound (262MB of gathered W). One wave per vocab
// entry; the 64x64 mat-vec runs on V_WMMA_F32_16X16X4_F32: A = 16x4 tile of
// W^T (sole real traffic, loaded once, coalesced), B = x chunk broadcast
// across the 16 columns. 64 WMMAs per vocab entry.
// ---------------------------------------------------------------------------
__global__ __launch_bounds__(256) void brillm_logits(
    const int* __restrict__ final_eids, const float* __restrict__ bias_table,
    const float* __restrict__ W, const float* __restrict__ bias,
    const float* __restrict__ x_in, float* __restrict__ out)
{
  __shared__ float xs[DD];
  __shared__ float bsum[8][DD];
  const int tid = threadIdx.x;
  if (tid < DD) xs[tid] = x_in[tid];

  const int wave = tid >> 5;
  const int lane = tid & 31;
  const int v = blockIdx.x * 8 + wave;         // grid = 2000 * 8 waves = 16000
  const int eid = final_eids[v];
  const float* __restrict__ Wv = W + (size_t)eid * (DD * DD);
  {
    const size_t be = (size_t)eid * DD, bv = (size_t)v * DD;
    bsum[wave][lane]      = bias[be + lane]      + bias_table[bv + lane];
    bsum[wave][lane + 32] = bias[be + lane + 32] + bias_table[bv + lane + 32];
  }
  __syncthreads();

  const int row = lane & 15;                   // A-matrix M index for this lane
  const int kb  = (lane & 16) ? 2 : 0;         // A/B K base for this half-wave
  float p = 0.f;

#if HAS_WMMA4
  v8f z = {0.f, 0.f, 0.f, 0.f, 0.f, 0.f, 0.f, 0.f};
  v8f cacc[4] = {z, z, z, z};                  // 4 blocks of 16 output dims
  for (int d0 = 0; d0 < DD; d0 += 4) {
    v2f b;                                     // B[k,n] = x[d0+k] (bcast over n)
    b.x = xs[d0 + kb];
    b.y = xs[d0 + kb + 1];
    const float* W0 = Wv + (size_t)(d0 + kb) * DD + row;
#pragma unroll
    for (int i = 0; i < 4; ++i) {
      v2f aop;                                 // A[m,k] = W[d0+k][i*16+m]
      aop.x = W0[i * 16];
      aop.y = W0[DD + i * 16];
      cacc[i] = __builtin_amdgcn_wmma_f32_16x16x4_f32(
          false, aop, false, b, (short)0, cacc[i], false, false);
    }
  }
  // C layout: vgpr r, lane l -> M = r + 8*(l>=16); columns redundant.
  const int mo = (lane & 16) ? 8 : 0;
#pragma unroll
  for (int i = 0; i < 4; ++i) {
#pragma unroll
    for (int r = 0; r < 8; ++r)
      p += tanhf(cacc[i][r] + bsum[wave][i * 16 + r + mo]);
  }
  p += __shfl_xor(p, 16, 32);                  // low half + high half of M
#else
  // scalar fallback: lane handles output dims lane and lane+32
  float a0 = 0.f, a1 = 0.f;
  for (int d = 0; d < DD; ++d) {
    const float xd = xs[d];
    a0 = fmaf(xd, Wv[d * DD + lane], a0);
    a1 = fmaf(xd, Wv[d * DD + lane + 32], a1);
  }
  p = tanhf(a0 + bsum[wave][lane]) + tanhf(a1 + bsum[wave][lane + 32]);
#pragma unroll
  for (int off = 16; off >= 1; off >>= 1) p += __shfl_xor(p, off, 32);
#endif
  if (lane == 0) out[v] = p;
}

extern "C" void kernel_launch(void* const* d_in, const int* in_sizes, int n_in,
                              void* d_out, int out_size, void* d_ws, size_t ws_size,
                              hipStream_t stream) {
  const int*   ids        = (const int*)  d_in[0];
  const int*   step_eids  = (const int*)  d_in[1];
  const int*   final_eids = (const int*)  d_in[2];
  const float* bias_table = (const float*)d_in[3];
  const float* W          = (const float*)d_in[4];
  const float* bias       = (const float*)d_in[5];
  const float* W_shared   = (const float*)d_in[6];
  const float* bias_sh    = (const float*)d_in[7];
  const float* a          = (const float*)d_in[8];
  const float* gate       = (const float*)d_in[9];
  const float* pe_scale   = (const float*)d_in[10];
  const float* PE         = (const float*)d_in[11];
  float* x_ws = (float*)d_ws;                 // 64 floats of scratch for x
  float* out  = (float*)d_out;

  hipLaunchKernelGGL(brillm_scan, dim3(1), dim3(64), 0, stream,
      ids, step_eids, bias_table, W, bias, W_shared, bias_sh, a, gate,
      pe_scale, PE, x_ws);
  hipLaunchKernelGGL(brillm_logits, dim3(VV / 8), dim3(256), 0, stream,
      final_eids, bias_table, W, bias, x_ws, out);
}